// ContrastiveLoss_3015067042304
// MI455X (gfx1250) — compile-verified
//
#include <hip/hip_runtime.h>

typedef float v2f __attribute__((ext_vector_type(2)));
typedef float v8f __attribute__((ext_vector_type(8)));

#define DIM      128
#define NUM_NEG  10
#define NLOGITS  (NUM_NEG + 1)
#define INV_TAU  (1.0f / 0.65f)

__global__ void zero_out_kernel(float* out) {
    if (threadIdx.x == 0) out[0] = 0.0f;
}

// One wave (32 lanes) handles a tile of 16 rows.
// WMMA diagonal trick: dot(e_m, v_m) for 16 rows at once = diag(A x B),
// A = e rows (16x128 f32), B col m = gathered vector of row m (128x16 f32),
// accumulated as 32 chained V_WMMA_F32_16X16X4_F32 (exact f32 math).
__global__ __launch_bounds__(256) void contrastive_loss_kernel(
    const float* __restrict__ emb,      // (n, 128)
    const float* __restrict__ negemb,   // (m, 128)
    const int*   __restrict__ posopt,   // (n, kcols)
    const int*   __restrict__ negidx,   // (NUM_NEG, n)
    const int*   __restrict__ iter_n,   // scalar on device
    float*       __restrict__ out,      // 1 float (pre-zeroed)
    int n, int kcols)
{
    const int lane   = threadIdx.x & 31;
    const int wave   = threadIdx.x >> 5;
    const int ntiles = (n + 15) >> 4;
    const int tile   = blockIdx.x * (blockDim.x >> 5) + wave;
    if (tile >= ntiles) return;           // whole wave exits uniformly

    const int  rowInTile = lane & 15;     // matrix row m this lane feeds
    int        r         = tile * 16 + rowInTile;
    const bool validRow  = (r < n);
    if (r >= n) r = n - 1;                // clamp; contribution zeroed later
    const int  colOff    = (lane >> 4) * 2;   // K sub-offset per WMMA layout

    // ---- Preload A tile: lane l holds e[r_{l&15}, 4c + colOff + {0,1}] ----
    const float* arow = emb + (size_t)r * DIM + colOff;
    v2f A[32];
    #pragma unroll
    for (int c = 0; c < 32; ++c)
        A[c] = *(const v2f*)(arow + 4 * c);

    // ---- Gather row pointers: j=0 positive, j=1..10 negatives ----
    const int col  = iter_n[0] - 1;
    const int pidx = posopt[(size_t)r * kcols + col];
    const float* brow[NLOGITS];
    brow[0] = emb + (size_t)pidx * DIM + colOff;
    #pragma unroll
    for (int j = 0; j < NUM_NEG; ++j) {
        const int ni = negidx[(size_t)j * n + r];
        brow[j + 1] = negemb + (size_t)ni * DIM + colOff;
    }
    #pragma unroll
    for (int j = 0; j < NLOGITS; ++j)
        __builtin_prefetch(brow[j], 0, 0);   // global_prefetch_b8

    float logit[NLOGITS];

    #pragma unroll
    for (int j = 0; j < NLOGITS; ++j) {
        const float* q = brow[j];
        v8f acc = {};
        #pragma unroll
        for (int c = 0; c < 32; ++c) {
            const v2f b = *(const v2f*)(q + 4 * c);
            // D = A(16x4) x B(4x16) + C ; 8 args (neg_a, A, neg_b, B, c_mod, C, reuse_a, reuse_b)
            acc = __builtin_amdgcn_wmma_f32_16x16x4_f32(
                false, A[c], false, b, (short)0, acc, false, false);
        }
        // Diagonal of 16x16 f32 D:
        //   m = 0..7  lives in lane m      at acc[m]
        //   m = 8..15 lives in lane m+16   at acc[m-8]
        const int sel = (lane < 16) ? lane : (lane - 24);  // acc index (0..7 where used)
        float dv = acc[0];
        dv = (sel == 1) ? acc[1] : dv;
        dv = (sel == 2) ? acc[2] : dv;
        dv = (sel == 3) ? acc[3] : dv;
        dv = (sel == 4) ? acc[4] : dv;
        dv = (sel == 5) ? acc[5] : dv;
        dv = (sel == 6) ? acc[6] : dv;
        dv = (sel == 7) ? acc[7] : dv;
        const int src = (rowInTile < 8) ? rowInTile : (rowInTile + 16);
        const float dot = __shfl(dv, src, 32);   // lane m (0..15) gets D[m,m]
        logit[j] = dot * INV_TAU;
    }

    // ---- logsumexp over 11 logits, minus positive logit (lanes 0..15) ----
    float mx = logit[0];
    #pragma unroll
    for (int j = 1; j < NLOGITS; ++j) mx = fmaxf(mx, logit[j]);
    float s = 0.0f;
    #pragma unroll
    for (int j = 0; j < NLOGITS; ++j) s += expf(logit[j] - mx);
    float score = mx + logf(s) - logit[0];
    if (lane >= 16 || !validRow) score = 0.0f;

    // ---- wave32 reduce + single atomic per wave ----
    #pragma unroll
    for (int off = 16; off > 0; off >>= 1)
        score += __shfl_xor(score, off, 32);
    if (lane == 0)
        atomicAdd(out, score / (float)n);
}

extern "C" void kernel_launch(void* const* d_in, const int* in_sizes, int n_in,
                              void* d_out, int out_size, void* d_ws, size_t ws_size,
                              hipStream_t stream) {
    const float* emb    = (const float*)d_in[0];
    const float* negemb = (const float*)d_in[1];
    const int*   posopt = (const int*)d_in[2];
    const int*   negidx = (const int*)d_in[3];
    const int*   iter_n = (const int*)d_in[4];
    float*       out    = (float*)d_out;

    const int n     = in_sizes[0] / DIM;       // 200000
    const int kcols = in_sizes[2] / n;         // 5

    zero_out_kernel<<<1, 32, 0, stream>>>(out);

    const int ntiles          = (n + 15) / 16;
    const int waves_per_block = 8;             // 256 threads = 8 wave32
    const int grid            = (ntiles + waves_per_block - 1) / waves_per_block;
    contrastive_loss_kernel<<<grid, 256, 0, stream>>>(
        emb, negemb, posopt, negidx, iter_n, out, n, kcols);
}